// SwarmBrain_11854109737383
// MI455X (gfx1250) — compile-verified
//
#include <hip/hip_runtime.h>
#include <math.h>

typedef float v2f __attribute__((ext_vector_type(2)));
typedef float v8f __attribute__((ext_vector_type(8)));

#define TPB 256

__global__ void zero_kernel(float* __restrict__ p, long n) {
  long i = (long)blockIdx.x * blockDim.x + threadIdx.x;
  long s = (long)gridDim.x * blockDim.x;
  for (; i < n; i += s) p[i] = 0.0f;
}

__global__ void deg_kernel(const int* __restrict__ col, float* __restrict__ deg, int E) {
  int i = blockIdx.x * blockDim.x + threadIdx.x;
  int s = gridDim.x * blockDim.x;
  for (; i < E; i += s) atomicAdd(&deg[col[i]], 1.0f);
}

__global__ void dis_kernel(const float* __restrict__ deg, float* __restrict__ dis, int N) {
  int i = blockIdx.x * blockDim.x + threadIdx.x;
  int s = gridDim.x * blockDim.x;
  for (; i < N; i += s) {
    float d = deg[i];
    dis[i] = (d > 0.0f) ? rsqrtf(fmaxf(d, 1.0f)) : 0.0f;
  }
}

__global__ void norm_kernel(const int* __restrict__ row, const int* __restrict__ col,
                            const float* __restrict__ dis, float* __restrict__ nrm, int E) {
  int i = blockIdx.x * blockDim.x + threadIdx.x;
  int s = gridDim.x * blockDim.x;
  for (; i < E; i += s) nrm[i] = dis[row[i]] * dis[col[i]];
}

// Layer 1: [N,5] @ [5,32] — tiny, plain VALU.
__global__ void gemm1_kernel(const float* __restrict__ x, const float* __restrict__ W1,
                             float* __restrict__ T, int N) {
  __shared__ float Ws[5 * 32];
  for (int i = threadIdx.x; i < 5 * 32; i += blockDim.x) Ws[i] = W1[i];
  __syncthreads();
  int n = blockIdx.x * blockDim.x + threadIdx.x;
  int s = gridDim.x * blockDim.x;
  for (; n < N; n += s) {
    float xv[5];
#pragma unroll
    for (int k = 0; k < 5; k++) xv[k] = x[n * 5 + k];
#pragma unroll
    for (int j = 0; j < 32; j++) {
      float acc = 0.0f;
#pragma unroll
      for (int k = 0; k < 5; k++) acc = fmaf(xv[k], Ws[k * 32 + j], acc);
      T[n * 32 + j] = acc;
    }
  }
}

// Layers 2/3: T = relu(Araw + bias) @ W, via V_WMMA_F32_16X16X4_F32.
// Wave handles a 16-node tile: K=32 in 8 k-steps, two 16-col halves.
// Full tiles take an unpredicated fast path (no exec save/restore around loads).
__global__ void gemm_wmma_kernel(const float* __restrict__ Araw, const float* __restrict__ bias,
                                 const float* __restrict__ W, float* __restrict__ T, int N) {
  const int lane  = threadIdx.x & 31;
  const int l16   = lane & 15;
  const int khalf = lane >> 4;   // 0: K+{0,1}, 1: K+{2,3} per ISA A/B f32 layout

  // Tile-invariant fragments: B (32x32 weight, row-major W[k][n]) and bias.
  v2f bfrag[8][2];
  v2f biasf[8];
#pragma unroll
  for (int kk = 0; kk < 8; kk++) {
    int k0 = kk * 4 + khalf * 2;
    biasf[kk].x = bias[k0 + 0];
    biasf[kk].y = bias[k0 + 1];
#pragma unroll
    for (int nh = 0; nh < 2; nh++) {
      bfrag[kk][nh].x = W[(k0 + 0) * 32 + nh * 16 + l16];
      bfrag[kk][nh].y = W[(k0 + 1) * 32 + nh * 16 + l16];
    }
  }

  int ntiles  = (N + 15) >> 4;
  int wave    = (int)((blockIdx.x * blockDim.x + threadIdx.x) >> 5);
  int wstride = (int)((gridDim.x * blockDim.x) >> 5);

  for (int tile = wave; tile < ntiles; tile += wstride) {
    const int base = tile * 16;
    v8f c0 = {0.f, 0.f, 0.f, 0.f, 0.f, 0.f, 0.f, 0.f};
    v8f c1 = {0.f, 0.f, 0.f, 0.f, 0.f, 0.f, 0.f, 0.f};

    if (base + 16 <= N) {
      // ---- fast path: all 16 rows valid, no per-lane predication ----
      const float* Ap = Araw + (size_t)(base + l16) * 32;
      v2f a[8];
#pragma unroll
      for (int kk = 0; kk < 8; kk++) {
        int kidx = kk * 4 + khalf * 2;
        v2f raw = *(const v2f*)(Ap + kidx);         // 8B-aligned b64 load
        a[kk].x = fmaxf(raw.x + biasf[kk].x, 0.0f);
        a[kk].y = fmaxf(raw.y + biasf[kk].y, 0.0f);
      }
#pragma unroll
      for (int kk = 0; kk < 8; kk++) {
        c0 = __builtin_amdgcn_wmma_f32_16x16x4_f32(false, a[kk], false, bfrag[kk][0],
                                                   (short)0, c0, false, false);
        c1 = __builtin_amdgcn_wmma_f32_16x16x4_f32(false, a[kk], false, bfrag[kk][1],
                                                   (short)0, c1, false, false);
      }
      // C/D layout: lanes 0-15 -> M=r, lanes 16-31 -> M=r+8; N = l16 (+16 for c1).
#pragma unroll
      for (int r = 0; r < 8; r++) {
        int mm = base + r + khalf * 8;
        T[mm * 32 + l16]      = c0[r];
        T[mm * 32 + 16 + l16] = c1[r];
      }
    } else {
      // ---- tail path (never taken when N % 16 == 0) ----
      int m = base + l16;
      bool valid = m < N;
#pragma unroll
      for (int kk = 0; kk < 8; kk++) {
        int kidx = kk * 4 + khalf * 2;
        v2f a;
        if (valid) {
          float ax = Araw[(size_t)m * 32 + kidx + 0] + biasf[kk].x;
          float ay = Araw[(size_t)m * 32 + kidx + 1] + biasf[kk].y;
          a.x = ax > 0.0f ? ax : 0.0f;
          a.y = ay > 0.0f ? ay : 0.0f;
        } else {
          a.x = 0.0f; a.y = 0.0f;
        }
        c0 = __builtin_amdgcn_wmma_f32_16x16x4_f32(false, a, false, bfrag[kk][0],
                                                   (short)0, c0, false, false);
        c1 = __builtin_amdgcn_wmma_f32_16x16x4_f32(false, a, false, bfrag[kk][1],
                                                   (short)0, c1, false, false);
      }
#pragma unroll
      for (int r = 0; r < 8; r++) {
        int mm = base + r + khalf * 8;
        if (mm < N) {
          T[mm * 32 + l16]      = c0[r];
          T[mm * 32 + 16 + l16] = c1[r];
        }
      }
    }
  }
}

// Edge scatter: wave per edge, lane = feature. agg[col] += norm * T[row].
__global__ void scatter_kernel(const float* __restrict__ T, const int* __restrict__ row,
                               const int* __restrict__ col, const float* __restrict__ nrm,
                               float* __restrict__ agg, int E) {
  int gid  = blockIdx.x * blockDim.x + threadIdx.x;
  int lane = gid & 31;
  int e    = gid >> 5;
  int s    = (gridDim.x * blockDim.x) >> 5;
  for (; e < E; e += s) {
    int r = row[e], c = col[e];
    float v = nrm[e] * T[r * 32 + lane];
    atomicAdd(&agg[c * 32 + lane], v);
  }
}

// Per-node heads: h = relu(agg+b3); out[n]=h.Wd+bd; out[N+n]=h.Wa+ba
__global__ void scores_kernel(const float* __restrict__ agg, const float* __restrict__ b3,
                              const float* __restrict__ Wd, const float* __restrict__ bd,
                              const float* __restrict__ Wa, const float* __restrict__ ba,
                              float* __restrict__ out, int N) {
  int gid  = blockIdx.x * blockDim.x + threadIdx.x;
  int lane = gid & 31;
  int n    = gid >> 5;
  int s    = (gridDim.x * blockDim.x) >> 5;
  for (; n < N; n += s) {
    float h = agg[n * 32 + lane] + b3[lane];
    h = h > 0.0f ? h : 0.0f;
    float pd = h * Wd[lane];
    float pa = h * Wa[lane];
#pragma unroll
    for (int off = 16; off > 0; off >>= 1) {
      pd += __shfl_xor(pd, off, 32);
      pa += __shfl_xor(pa, off, 32);
    }
    if (lane == 0) {
      out[n]     = pd + bd[0];
      out[N + n] = pa + ba[0];
    }
  }
}

// Single-block argmax (first-max tie break, like jnp.argmax) + tiny head GEMVs.
__global__ void head_kernel(const float* __restrict__ ascore, const float* __restrict__ agg,
                            const float* __restrict__ b3,
                            const float* __restrict__ Wt, const float* __restrict__ bt,
                            const float* __restrict__ Wact, const float* __restrict__ bact,
                            float* __restrict__ outh, int N) {
  __shared__ float sval[1024];
  __shared__ int   sidx[1024];
  int tid = threadIdx.x;
  float bv = -3.402823466e38f;
  int   bi = 0x7fffffff;
  for (int i = tid; i < N; i += blockDim.x) {
    float v = ascore[i];
    if (v > bv) { bv = v; bi = i; }  // i strictly increases -> first max kept
  }
  sval[tid] = bv; sidx[tid] = bi;
  __syncthreads();
  for (int off = blockDim.x >> 1; off > 0; off >>= 1) {
    if (tid < off) {
      float ov = sval[tid + off]; int oi = sidx[tid + off];
      if (ov > sval[tid] || (ov == sval[tid] && oi < sidx[tid])) {
        sval[tid] = ov; sidx[tid] = oi;
      }
    }
    __syncthreads();
  }
  int idx = sidx[0];
  if (tid < 11) {
    if (tid < 2) {
      float acc = bt[tid];
#pragma unroll
      for (int k = 0; k < 32; k++) {
        float h = agg[idx * 32 + k] + b3[k]; h = h > 0.0f ? h : 0.0f;
        acc = fmaf(h, Wt[k * 2 + tid], acc);
      }
      outh[tid] = acc;
    } else {
      int j = tid - 2;
      float acc = bact[j];
#pragma unroll
      for (int k = 0; k < 32; k++) {
        float h = agg[idx * 32 + k] + b3[k]; h = h > 0.0f ? h : 0.0f;
        acc = fmaf(h, Wact[k * 9 + j], acc);
      }
      outh[2 + j] = acc;
    }
  }
}

extern "C" void kernel_launch(void* const* d_in, const int* in_sizes, int n_in,
                              void* d_out, int out_size, void* d_ws, size_t ws_size,
                              hipStream_t stream) {
  const float* x    = (const float*)d_in[0];
  const int*   ei   = (const int*)  d_in[1];
  const float* W1   = (const float*)d_in[2];
  const float* b1   = (const float*)d_in[3];
  const float* W2   = (const float*)d_in[4];
  const float* b2   = (const float*)d_in[5];
  const float* W3   = (const float*)d_in[6];
  const float* b3   = (const float*)d_in[7];
  const float* Wd   = (const float*)d_in[8];
  const float* bd   = (const float*)d_in[9];
  const float* Wa   = (const float*)d_in[10];
  const float* ba   = (const float*)d_in[11];
  const float* Wt   = (const float*)d_in[12];
  const float* bt   = (const float*)d_in[13];
  const float* Wact = (const float*)d_in[14];
  const float* bact = (const float*)d_in[15];

  const int N = in_sizes[0] / 5;
  const int E = in_sizes[1] / 2;
  const int* row = ei;
  const int* col = ei + E;

  float* ws  = (float*)d_ws;
  float* deg = ws;                       // N
  float* dis = deg + N;                  // N
  float* nrm = dis + N;                  // E
  float* T   = nrm + E;                  // 32N (GEMM outputs, reused per layer)
  float* A   = T + 32 * (size_t)N;       // 32N (agg1, then agg3)
  float* B   = A + 32 * (size_t)N;       // 32N (agg2)

  float* out = (float*)d_out;

  const int grid_e  = (E + TPB - 1) / TPB;          // thread-per-edge kernels
  const int grid_n  = (N + TPB - 1) / TPB;          // thread-per-node kernels
  const int grid_ew = (E + (TPB / 32) - 1) / (TPB / 32);  // wave-per-edge
  const int grid_nw = (N + (TPB / 32) - 1) / (TPB / 32);  // wave-per-node
  const int grid_f  = (int)((32L * N + TPB - 1) / TPB);   // feature-buffer zero
  const int ntiles  = (N + 15) / 16;
  const int grid_g  = (ntiles + (TPB / 32) - 1) / (TPB / 32);

  // Normalization: deg -> dis -> norm (recomputed every call; ws not re-poisoned).
  zero_kernel<<<grid_n, TPB, 0, stream>>>(deg, N);
  deg_kernel<<<grid_e, TPB, 0, stream>>>(col, deg, E);
  dis_kernel<<<grid_n, TPB, 0, stream>>>(deg, dis, N);
  norm_kernel<<<grid_e, TPB, 0, stream>>>(row, col, dis, nrm, E);

  // Layer 1: T = x @ W1 ; A = scatter(T) ; (bias+relu fused into next GEMM's A-load)
  gemm1_kernel<<<grid_n, TPB, 0, stream>>>(x, W1, T, N);
  zero_kernel<<<grid_f, TPB, 0, stream>>>(A, 32L * N);
  scatter_kernel<<<grid_ew, TPB, 0, stream>>>(T, row, col, nrm, A, E);

  // Layer 2: T = relu(A + b1) @ W2 ; B = scatter(T)
  gemm_wmma_kernel<<<grid_g, TPB, 0, stream>>>(A, b1, W2, T, N);
  zero_kernel<<<grid_f, TPB, 0, stream>>>(B, 32L * N);
  scatter_kernel<<<grid_ew, TPB, 0, stream>>>(T, row, col, nrm, B, E);

  // Layer 3: T = relu(B + b2) @ W3 ; A = scatter(T)  (A now holds agg3)
  gemm_wmma_kernel<<<grid_g, TPB, 0, stream>>>(B, b2, W3, T, N);
  zero_kernel<<<grid_f, TPB, 0, stream>>>(A, 32L * N);
  scatter_kernel<<<grid_ew, TPB, 0, stream>>>(T, row, col, nrm, A, E);

  // Heads: per-node scores, then argmax + target/action heads.
  scores_kernel<<<grid_nw, TPB, 0, stream>>>(A, b3, Wd, bd, Wa, ba, out, N);
  head_kernel<<<1, 1024, 0, stream>>>(out + N, A, b3, Wt, bt, Wact, bact, out + 2 * (size_t)N, N);
}